// DistGraphConv_51032801411438
// MI455X (gfx1250) — compile-verified
//
#include <hip/hip_runtime.h>

#define N_NODES 50000
#define N_EDGES 800000
#define FEATS   128
#define M_TILES (N_NODES / 16)   // 3125 (exact)
#define N_TILES (FEATS / 16)     // 8

typedef __attribute__((ext_vector_type(2))) float v2f;
typedef __attribute__((ext_vector_type(8))) float v8f;

__device__ __forceinline__ void atomic_add_f32(float* p, float v) {
    __hip_atomic_fetch_add(p, v, __ATOMIC_RELAXED, __HIP_MEMORY_SCOPE_AGENT);
}

// ---------------- zero fill ----------------
__global__ void zero_kernel(float* __restrict__ p, int n) {
    int i = blockIdx.x * blockDim.x + threadIdx.x;
    if (i < n) p[i] = 0.0f;
}

// ---------------- degree counting ----------------
__global__ void degree_kernel(const int* __restrict__ src, const int* __restrict__ dst,
                              float* __restrict__ deg_out, float* __restrict__ deg_in) {
    int e = blockIdx.x * blockDim.x + threadIdx.x;
    if (e < N_EDGES) {
        atomic_add_f32(&deg_out[src[e]], 1.0f);
        atomic_add_f32(&deg_in[dst[e]], 1.0f);
    }
}

// ---------------- deg -> rsqrt(max(deg,1)) in place ----------------
__global__ void rsqrt_kernel(float* __restrict__ deg_out, float* __restrict__ deg_in) {
    int i = blockIdx.x * blockDim.x + threadIdx.x;
    if (i < N_NODES) {
        deg_out[i] = rsqrtf(fmaxf(deg_out[i], 1.0f));
        deg_in[i]  = rsqrtf(fmaxf(deg_in[i],  1.0f));
    }
}

// ---------------- y = (x * s_out[row]) @ W  via V_WMMA_F32_16X16X4_F32 ----------------
// One wave per 16x16 output tile; K=128 -> 32 wmma steps.
// A layout (16x4 f32, ISA 7.12.2): lane l: m=l&15, half h=l>>4; VGPR{0,1} = A[m][k0+2h+{0,1}]
// B layout (4x16 f32, symmetric): VGPR{0,1} = B[k0+2h+{0,1}][n], n=l&15
// D layout: VGPR j -> row (16*tile_m + j + 8h), col (16*tile_n + (l&15))
__global__ __launch_bounds__(256)
void gemm_wmma_kernel(const float* __restrict__ x, const float* __restrict__ W,
                      const float* __restrict__ s_out, float* __restrict__ y) {
    const int wave   = (blockIdx.x * blockDim.x + threadIdx.x) >> 5;   // 0..24999
    const int lane   = threadIdx.x & 31;
    const int tile_n = wave & (N_TILES - 1);
    const int tile_m = wave >> 3;

    const int m = lane & 15;
    const int h = lane >> 4;
    const int row = tile_m * 16 + m;          // A row for this lane
    const int col = tile_n * 16 + m;          // B column for this lane

    const float scale = s_out[row];           // fused left-norm
    const float* __restrict__ arow = x + (size_t)row * FEATS;

    v8f c = {};
    #pragma unroll 4
    for (int k0 = 0; k0 < FEATS; k0 += 4) {
        const int k = k0 + 2 * h;
        v2f a = *(const v2f*)(arow + k);
        a.x *= scale;
        a.y *= scale;
        v2f b;
        b.x = W[(size_t)k * FEATS + col];
        b.y = W[(size_t)(k + 1) * FEATS + col];
        c = __builtin_amdgcn_wmma_f32_16x16x4_f32(
                /*neg_a=*/false, a, /*neg_b=*/false, b,
                /*c_mod=*/(short)0, c, /*reuse_a=*/false, /*reuse_b=*/false);
    }

    #pragma unroll
    for (int j = 0; j < 8; ++j) {
        const int orow = tile_m * 16 + j + 8 * h;
        y[(size_t)orow * FEATS + col] = c[j];
    }
}

// ---------------- out[dst] += y[src]  (one wave per edge, float4 per lane) ----------------
__global__ __launch_bounds__(256)
void scatter_kernel(const int* __restrict__ src, const int* __restrict__ dst,
                    const float* __restrict__ y, float* __restrict__ out) {
    const int e    = (blockIdx.x * blockDim.x + threadIdx.x) >> 5;
    const int lane = threadIdx.x & 31;
    if (e >= N_EDGES) return;
    const int s = src[e];
    const int d = dst[e];
    const float4 v = *(const float4*)(y + (size_t)s * FEATS + lane * 4);
    float* o = out + (size_t)d * FEATS + lane * 4;
    atomic_add_f32(o + 0, v.x);
    atomic_add_f32(o + 1, v.y);
    atomic_add_f32(o + 2, v.z);
    atomic_add_f32(o + 3, v.w);
}

// ---------------- out = out * s_in[row] + bias[col] ----------------
__global__ void finalize_kernel(float* __restrict__ out, const float* __restrict__ s_in,
                                const float* __restrict__ bias) {
    int i = blockIdx.x * blockDim.x + threadIdx.x;
    if (i < N_NODES * FEATS) {
        int r = i >> 7;
        int ccol = i & (FEATS - 1);
        out[i] = out[i] * s_in[r] + bias[ccol];
    }
}

extern "C" void kernel_launch(void* const* d_in, const int* in_sizes, int n_in,
                              void* d_out, int out_size, void* d_ws, size_t ws_size,
                              hipStream_t stream) {
    const float* x    = (const float*)d_in[0];
    const int*   src  = (const int*)  d_in[1];
    const int*   dst  = (const int*)  d_in[2];
    const float* W    = (const float*)d_in[3];
    const float* bias = (const float*)d_in[4];
    float* out = (float*)d_out;

    // workspace layout
    float* s_out = (float*)d_ws;                       // N_NODES floats
    float* s_in  = s_out + N_NODES;                    // N_NODES floats
    float* y     = s_in + N_NODES;                     // N_NODES*FEATS floats (offset 400000 B, 16B aligned)

    const int TB = 256;

    // zero degree buffers and output accumulator
    zero_kernel<<<(2 * N_NODES + TB - 1) / TB, TB, 0, stream>>>(s_out, 2 * N_NODES);
    zero_kernel<<<(N_NODES * FEATS + TB - 1) / TB, TB, 0, stream>>>(out, N_NODES * FEATS);

    // degrees
    degree_kernel<<<(N_EDGES + TB - 1) / TB, TB, 0, stream>>>(src, dst, s_out, s_in);

    // rsqrt(clamp(deg,1))
    rsqrt_kernel<<<(N_NODES + TB - 1) / TB, TB, 0, stream>>>(s_out, s_in);

    // y = (x * s_out) @ W   — 25000 waves, 8 waves per block
    gemm_wmma_kernel<<<(M_TILES * N_TILES) / 8, TB, 0, stream>>>(x, W, s_out, y);

    // out[dst] += y[src]
    scatter_kernel<<<(N_EDGES * 32) / TB, TB, 0, stream>>>(src, dst, y, out);

    // out = out * s_in + bias
    finalize_kernel<<<(N_NODES * FEATS + TB - 1) / TB, TB, 0, stream>>>(out, s_in, bias);
}